// ConnectionLoss_41729902248394
// MI455X (gfx1250) — compile-verified
//
#include <hip/hip_runtime.h>
#include <math.h>

// Problem geometry: pred [32, 1, 512, 512] float32 -> scalar float loss.
#define NPIX      262144        // 512*512 (power of two)
#define NIMG      32
#define NTOT      8388608       // 32 * 262144
#define IMG_SHIFT 18            // g >> 18 = image index
#define ROW_SHIFT 9             // 512 columns
#define WMASK     511
#define HWMASK    262143
#define INFI      0x7FFFFFFF
#define NEGC      (-100.0f)

// per-image scalar slots
#define SC_C  0   // number of roots (components)
#define SC_RG 1   // max root index (-1 if no fg)
#define SC_MN 2   // min fg index >= 1 (INFI if none)
#define SC_P0 3   // the "v" value from the reference quirk
#define SC_RV 4   // selected root (target = root==rv); -2 => target all zero
#define SC_STRIDE 8

typedef float v2f __attribute__((ext_vector_type(2)));
typedef float v8f __attribute__((ext_vector_type(8)));

// ---------------- lock-free union-find (min-index roots) ----------------
__device__ __forceinline__ int uf_find(int* L, int x) {
  int p = L[x];
  while (p != x) { x = p; p = L[x]; }   // parents strictly decrease -> terminates
  return x;
}

__device__ __forceinline__ void uf_union(int* L, int a, int b) {
  while (true) {
    a = uf_find(L, a);
    b = uf_find(L, b);
    if (a == b) return;
    int hi = a > b ? a : b;
    int lo = a ^ b ^ hi;
    int old = atomicMin(&L[hi], lo);    // link larger root toward smaller index
    if (old == hi) return;              // we installed the link
    a = old; b = lo;                    // raced: retry with observed parent
  }
}

// ---------------- kernels ----------------
__global__ void k_init_scalars(int* scal) {
  int n = threadIdx.x;
  if (n < NIMG) {
    int* s = scal + n * SC_STRIDE;
    s[SC_C] = 0; s[SC_RG] = -1; s[SC_MN] = INFI; s[SC_P0] = 0; s[SC_RV] = -2;
  }
}

__global__ void k_init_labels(const float* __restrict__ pred, int* __restrict__ L) {
  int g = blockIdx.x * blockDim.x + threadIdx.x;
  int hw = g & HWMASK;                  // per-image local flat index
  L[g] = (pred[g] >= 0.5f) ? hw : -1;   // -1 marks background
}

__global__ void k_merge(int* L) {
  int g = blockIdx.x * blockDim.x + threadIdx.x;
  int hw = g & HWMASK;
  int* Lb = L + (g - hw);               // image base
  if (Lb[hw] < 0) return;
  int w = hw & WMASK;
  int h = hw >> ROW_SHIFT;
  if (w > 0 && Lb[hw - 1]   >= 0) uf_union(Lb, hw, hw - 1);
  if (h > 0 && Lb[hw - 512] >= 0) uf_union(Lb, hw, hw - 512);
}

__global__ void k_compress(int* L) {
  int g = blockIdx.x * blockDim.x + threadIdx.x;
  int hw = g & HWMASK;
  int* Lb = L + (g - hw);
  if (Lb[hw] >= 0) Lb[hw] = uf_find(Lb, hw);   // final root = min index of component
}

// per 1024-pixel block: root count (for the rank scan) + per-image aggregates
__global__ void k_stats(const int* __restrict__ L, int* __restrict__ blockCnt, int* scal) {
  __shared__ int s_cnt, s_rg, s_mn;
  int t = threadIdx.x;
  if (t == 0) { s_cnt = 0; s_rg = -1; s_mn = INFI; }
  __syncthreads();
  int g = blockIdx.x * 1024 + t;
  int n = g >> IMG_SHIFT;
  int hw = g & HWMASK;
  int lab = L[g];
  if (lab >= 0) {
    if (lab == hw) atomicAdd(&s_cnt, 1);      // root pixel
    atomicMax(&s_rg, lab);
    if (hw >= 1) atomicMin(&s_mn, hw);
  }
  __syncthreads();
  if (t == 0) {
    blockCnt[blockIdx.x] = s_cnt;
    int* s = scal + n * SC_STRIDE;
    if (s_cnt) atomicAdd(&s[SC_C], s_cnt);
    if (s_rg >= 0) atomicMax(&s[SC_RG], s_rg);
    if (s_mn < INFI) atomicMin(&s[SC_MN], s_mn);
  }
}

// one block per image: exclusive scan of the 256 block root-counts + compute p0
__global__ void k_scan(const int* __restrict__ blockCnt, int* __restrict__ blockPre, int* scal) {
  __shared__ int sh[256];
  int n = blockIdx.x, t = threadIdx.x;
  int b = n * 256 + t;
  int own = blockCnt[b];
  sh[t] = own;
  __syncthreads();
  for (int off = 1; off < 256; off <<= 1) {
    int v = sh[t];
    int a = (t >= off) ? sh[t - off] : 0;
    __syncthreads();
    sh[t] = v + a;
    __syncthreads();
  }
  blockPre[b] = sh[t] - own;                   // exclusive prefix of root counts
  if (t == 0) {
    int* s = scal + n * SC_STRIDE;
    int mn = s[SC_MN], rg = s[SC_RG];
    int p0;
    if (mn == INFI)   p0 = 1;    // no fg pixel at index>=1: argmax of all-equal -> v=1
    else if (rg >= 1) p0 = rg;   // first pixel of last component IS its root index
    else              p0 = mn;   // single component rooted at pixel 0
    s[SC_P0] = p0;
  }
}

// find the p0-th smallest root (only when p0 <= C); writes SC_RV
__global__ void k_select(const int* __restrict__ L, const int* __restrict__ blockPre, int* scal) {
  int b = blockIdx.x;
  int n = b >> 8;                                // 256 blocks per image
  int p0 = scal[n * SC_STRIDE + SC_P0];
  int C  = scal[n * SC_STRIDE + SC_C];
  if (p0 > C) return;                            // target is all-zero; RV stays -2
  __shared__ int sh[1024];
  int t = threadIdx.x;
  int g = b * 1024 + t;
  int hw = g & HWMASK;
  int lab = L[g];
  int isR = (lab >= 0 && lab == hw) ? 1 : 0;
  sh[t] = isR;
  __syncthreads();
  for (int off = 1; off < 1024; off <<= 1) {     // inclusive Hillis-Steele scan
    int v = sh[t];
    int a = (t >= off) ? sh[t - off] : 0;
    __syncthreads();
    sh[t] = v + a;
    __syncthreads();
  }
  int rank = blockPre[b] + sh[t];                // 1-based rank of this root
  if (isR && rank == p0) scal[n * SC_STRIDE + SC_RV] = hw;
}

// clamped BCE with target = (root == rv); wave32 reduction via V_WMMA_F32_16X16X4_F32
__global__ void k_bce(const float* __restrict__ pred, const int* __restrict__ L,
                      const int* __restrict__ scal, double* __restrict__ blockSums) {
  int b = blockIdx.x;
  int t = threadIdx.x;
  int n = b >> 8;
  int rv = scal[n * SC_STRIDE + SC_RV];          // -2 never matches any label
  int base = b * 1024;
  float s = 0.0f;
#pragma unroll
  for (int k = 0; k < 4; ++k) {
    int g = base + k * 256 + t;
    float p = pred[g];
    int lab = L[g];
    float term = (lab == rv) ? fmaxf(logf(p), NEGC)
                             : fmaxf(log1pf(-p), NEGC);
    s += term;
  }
  // A = [s, 0] per lane (16x4), B = ones (4x16)  =>  D[m][n] = s[m] + s[m+16].
  // Lanes 0-15 hold rows 0..7, lanes 16-31 rows 8..15; sum 8 regs + pairwise shfl
  // over lane^16 yields the exact f32 wave sum through the matrix pipe.
  v2f a;  a[0] = s;    a[1] = 0.0f;
  v2f bb; bb[0] = 1.0f; bb[1] = 1.0f;
  v8f c = {};
  c = __builtin_amdgcn_wmma_f32_16x16x4_f32(false, a, false, bb, (short)0, c, false, false);
  float r = c[0] + c[1] + c[2] + c[3] + c[4] + c[5] + c[6] + c[7];
  r += __shfl_xor(r, 16, 32);
  __shared__ float wsum[8];
  int lane = t & 31, wid = t >> 5;
  if (lane == 0) wsum[wid] = r;
  __syncthreads();
  if (t == 0) {
    double d = 0.0;
    for (int i = 0; i < 8; ++i) d += (double)wsum[i];
    blockSums[b] = d;                            // fixed slot: deterministic
  }
}

__global__ void k_final(const double* __restrict__ blockSums, float* __restrict__ out) {
  __shared__ double sd[256];
  int t = threadIdx.x;
  double d = 0.0;
  for (int j = 0; j < 32; ++j) d += blockSums[t + 256 * j];  // fixed order
  sd[t] = d;
  __syncthreads();
  for (int s = 128; s > 0; s >>= 1) {
    if (t < s) sd[t] += sd[t + s];
    __syncthreads();
  }
  if (t == 0) out[0] = (float)(-sd[0] / 8388608.0);
}

// ---------------- launch ----------------
// ws layout (bytes):
//   [0,      1024)   int  scal[32*8]
//   [8192,  40960)   int  blockCnt[8192]
//   [40960, 73728)   int  blockPre[8192]
//   [81920, 147456)  dbl  blockSums[8192]
//   [262144, 262144+33554432)  int L[NTOT]
extern "C" void kernel_launch(void* const* d_in, const int* in_sizes, int n_in,
                              void* d_out, int out_size, void* d_ws, size_t ws_size,
                              hipStream_t stream) {
  (void)in_sizes; (void)n_in; (void)out_size; (void)ws_size;
  const float* pred = (const float*)d_in[0];
  float* out = (float*)d_out;
  char* ws = (char*)d_ws;
  int*    scal      = (int*)(ws + 0);
  int*    blockCnt  = (int*)(ws + 8192);
  int*    blockPre  = (int*)(ws + 40960);
  double* blockSums = (double*)(ws + 81920);
  int*    L         = (int*)(ws + 262144);

  k_init_scalars<<<1, 64, 0, stream>>>(scal);
  k_init_labels <<<NTOT / 256, 256, 0, stream>>>(pred, L);
  k_merge       <<<NTOT / 256, 256, 0, stream>>>(L);
  k_compress    <<<NTOT / 256, 256, 0, stream>>>(L);
  k_stats       <<<NTOT / 1024, 1024, 0, stream>>>(L, blockCnt, scal);
  k_scan        <<<NIMG, 256, 0, stream>>>(blockCnt, blockPre, scal);
  k_select      <<<NTOT / 1024, 1024, 0, stream>>>(L, blockPre, scal);
  k_bce         <<<NTOT / 1024, 256, 0, stream>>>(pred, L, scal, blockSums);
  k_final       <<<1, 256, 0, stream>>>(blockSums, out);
}